// NMS_73804718014592
// MI455X (gfx1250) — compile-verified
//
#include <hip/hip_runtime.h>
#include <stdint.h>

#define NBOX   25200
#define BATCH  32
#define KSEL   1024
#define MAXDET 300
#define CONF_T 0.25f
#define IOU_T  0.45f

typedef __attribute__((ext_vector_type(2))) float v2f;
typedef __attribute__((ext_vector_type(8))) float v8f;

// monotonic float -> uint mapping (larger float => larger key)
__device__ __forceinline__ uint32_t f2ord(float s) {
  uint32_t u = __float_as_uint(s);
  return (u & 0x80000000u) ? ~u : (u | 0x80000000u);
}
__device__ __forceinline__ float ord2f(uint32_t u) {
  return (u & 0x80000000u) ? __uint_as_float(u ^ 0x80000000u) : __uint_as_float(~u);
}

// ---------------------------------------------------------------------------
// K0: zero output + mask, build composite sort keys
//     key = (ordered(score) << 32) | (0xFFFFFFFF - idx)  -> all distinct,
//     ordering == (score desc, idx asc) which matches jax top_k tie-breaking.
// ---------------------------------------------------------------------------
__global__ void k_init(const float* __restrict__ x, float* __restrict__ out,
                       unsigned long long* __restrict__ keys,
                       uint32_t* __restrict__ mask) {
  const int stride = gridDim.x * blockDim.x;
  const int g = blockIdx.x * blockDim.x + threadIdx.x;
  for (int i = g; i < BATCH * MAXDET * 6; i += stride) out[i] = 0.0f;
  for (int i = g; i < BATCH * KSEL * (KSEL / 32); i += stride) mask[i] = 0u;
  for (int i = g; i < BATCH * NBOX; i += stride) {
    const float* p = x + (size_t)i * 6;
    float conf = p[4] * p[5];
    float s = (conf > CONF_T) ? conf : -1.0f;
    uint32_t n = (uint32_t)(i % NBOX);
    keys[i] = ((unsigned long long)f2ord(s) << 32) |
              (unsigned long long)(0xFFFFFFFFu - n);
  }
}

// ---------------------------------------------------------------------------
// K1: per-batch exact top-1024 via 8-pass radix select on 64-bit keys,
//     compact + bitonic sort (descending) in LDS, gather xyxy boxes/scores.
// ---------------------------------------------------------------------------
__global__ __launch_bounds__(1024) void k_select(
    const float* __restrict__ x, const unsigned long long* __restrict__ keysAll,
    float* __restrict__ topScore, float4* __restrict__ topBox) {
  __shared__ uint32_t hist[256];
  __shared__ unsigned long long sel[KSEL];
  __shared__ unsigned long long sPrefix;
  __shared__ uint32_t sNeed, sCnt;

  const int b = blockIdx.x;
  const int tid = threadIdx.x;
  const unsigned long long* keys = keysAll + (size_t)b * NBOX;

  if (tid == 0) { sPrefix = 0ull; sNeed = KSEL; sCnt = 0u; }
  __syncthreads();

  for (int p = 7; p >= 0; --p) {
    if (tid < 256) hist[tid] = 0u;
    __syncthreads();
    const unsigned long long pref = sPrefix;
    const int shift = 8 * p;
    for (int n = tid; n < NBOX; n += 1024) {
      unsigned long long k = keys[n];
      bool match = (p == 7) || ((k >> (shift + 8)) == (pref >> (shift + 8)));
      if (match) atomicAdd(&hist[(uint32_t)((k >> shift) & 255ull)], 1u);
    }
    __syncthreads();
    if (tid == 0) {
      uint32_t need = sNeed;
      for (int d = 255; d >= 0; --d) {
        uint32_t c = hist[d];
        if (c >= need) {
          sPrefix = pref | ((unsigned long long)(uint32_t)d << shift);
          sNeed = need;
          break;
        }
        need -= c;
      }
    }
    __syncthreads();
  }

  // exactly 1024 keys are >= T (keys are distinct)
  const unsigned long long T = sPrefix;
  for (int n = tid; n < NBOX; n += 1024) {
    unsigned long long k = keys[n];
    if (k >= T) { uint32_t pos = atomicAdd(&sCnt, 1u); sel[pos & (KSEL - 1)] = k; }
  }
  __syncthreads();

  // bitonic sort, descending
  for (uint32_t k2 = 2; k2 <= KSEL; k2 <<= 1) {
    for (uint32_t j = k2 >> 1; j > 0; j >>= 1) {
      uint32_t i = (uint32_t)tid, ixj = i ^ j;
      if (ixj > i) {
        unsigned long long va = sel[i], vb = sel[ixj];
        bool desc = ((i & k2) == 0);
        if (desc ? (va < vb) : (va > vb)) { sel[i] = vb; sel[ixj] = va; }
      }
      __syncthreads();
    }
  }

  unsigned long long key = sel[tid];
  uint32_t n = 0xFFFFFFFFu - (uint32_t)(key & 0xFFFFFFFFull);
  float score = ord2f((uint32_t)(key >> 32));
  const float* p = x + ((size_t)b * NBOX + n) * 6;
  float cx = p[0], cy = p[1], w = p[2], h = p[3];
  float4 box;
  box.x = cx - w * 0.5f; box.y = cy - h * 0.5f;
  box.z = cx + w * 0.5f; box.w = cy + h * 0.5f;
  topBox[(size_t)b * KSEL + tid] = box;
  topScore[(size_t)b * KSEL + tid] = score;
}

// ---------------------------------------------------------------------------
// K2: IoU suppression bitmask. One wave32 per 16x16 tile of the 1024x1024
//     matrix. The (area_i + area_j) term is a rank-2 outer sum computed with
//     V_WMMA_F32_16X16X4_F32: A(16x4)=[area_i,1,0,0], B(4x16)=[1;area_j;0;0].
//     A layout: lanes 0-15 hold K=0,1 (M=lane); lanes 16-31 hold K=2,3 (=0).
//     C layout: lane -> (N=lane&15, M=v+8*(lane>>4)).
// ---------------------------------------------------------------------------
__global__ __launch_bounds__(256) void k_iou(const float4* __restrict__ topBox,
                                             uint32_t* __restrict__ mask) {
  const int lane = threadIdx.x & 31;
  const int waveId = blockIdx.x * (blockDim.x >> 5) + (threadIdx.x >> 5);
  const int b  = waveId >> 12;     // 4096 tiles per batch
  const int t  = waveId & 4095;
  const int ti = t >> 6;
  const int tj = t & 63;
  const int nLocal = lane & 15;
  const int half   = lane >> 4;

  const float4* boxes = topBox + (size_t)b * KSEL;

  float4 bj = boxes[tj * 16 + nLocal];                 // column box (N = nLocal)
  float areaJ = (bj.z - bj.x) * (bj.w - bj.y);
  float4 ba = boxes[ti * 16 + nLocal];                 // row box for A (M = nLocal)
  float areaA = (ba.z - ba.x) * (ba.w - ba.y);
  float4 bi[8];                                        // row boxes M = v + 8*half
#pragma unroll
  for (int v = 0; v < 8; ++v) bi[v] = boxes[ti * 16 + v + 8 * half];

  v2f a, bv;
  a.x  = half ? 0.0f : areaA;  a.y  = half ? 0.0f : 1.0f;   // A cols 0,1 (2,3 = 0)
  bv.x = half ? 0.0f : 1.0f;   bv.y = half ? 0.0f : areaJ;  // B rows 0,1 (2,3 = 0)
  v8f c = {};
  c = __builtin_amdgcn_wmma_f32_16x16x4_f32(false, a, false, bv,
                                            (short)0, c, false, false);
  // c[v] = area_row(ti*16 + v + 8*half) + area_col(tj*16 + nLocal)

  const int jG = tj * 16 + nLocal;
#pragma unroll
  for (int v = 0; v < 8; ++v) {
    const int iG = ti * 16 + v + 8 * half;
    float ltx = fmaxf(bi[v].x, bj.x);
    float lty = fmaxf(bi[v].y, bj.y);
    float rbx = fminf(bi[v].z, bj.z);
    float rby = fminf(bi[v].w, bj.w);
    float iw = fmaxf(rbx - ltx, 0.0f);
    float ih = fmaxf(rby - lty, 0.0f);
    float inter = iw * ih;
    float iou = inter / (c[v] - inter + 1e-9f);
    bool pred = (iou > IOU_T) && (jG > iG);
    uint32_t bal = __builtin_amdgcn_ballot_w32(pred);
    if (lane == 0) {
      uint32_t lo = bal & 0xFFFFu;   // rows M = v       (lanes 0-15)
      uint32_t hi = bal >> 16;       // rows M = v + 8   (lanes 16-31)
      int sh = (tj & 1) * 16;
      uint32_t w = (uint32_t)(tj >> 1);
      if (lo) atomicOr(&mask[((size_t)b * KSEL + (ti * 16 + v)) * 32 + w], lo << sh);
      if (hi) atomicOr(&mask[((size_t)b * KSEL + (ti * 16 + v + 8)) * 32 + w], hi << sh);
    }
  }
}

// ---------------------------------------------------------------------------
// K3: sequential greedy sweep (wave0, remv bitmask in LDS), then ordered
//     compaction of survivors (already score-descending) into out[b][0..299].
// ---------------------------------------------------------------------------
__global__ __launch_bounds__(1024) void k_nms(
    const float* __restrict__ topScore, const float4* __restrict__ topBox,
    const uint32_t* __restrict__ mask, float* __restrict__ out) {
  __shared__ uint32_t remv[32], keepw[32], wpre[32];
  const int b = blockIdx.x;
  const int tid = threadIdx.x;
  if (tid < 32) { remv[tid] = 0u; keepw[tid] = 0u; }
  __syncthreads();

  if (tid < 32) {
    const uint32_t lane = (uint32_t)tid;
    for (int i = 0; i < KSEL; ++i) {
      uint32_t w = (uint32_t)i >> 5, bit = (uint32_t)i & 31u;
      bool sup = (remv[w] >> bit) & 1u;
      float sc = topScore[(size_t)b * KSEL + i];
      if (!sup && sc > CONF_T) {
        remv[lane] |= mask[((size_t)b * KSEL + i) * 32 + lane];
        if (lane == w) keepw[w] |= (1u << bit);
      }
    }
  }
  __syncthreads();

  if (tid == 0) {
    uint32_t s = 0;
    for (int w = 0; w < 32; ++w) { wpre[w] = s; s += __popc(keepw[w]); }
  }
  __syncthreads();

  {
    const int i = tid;
    uint32_t w = (uint32_t)i >> 5, bit = (uint32_t)i & 31u;
    if ((keepw[w] >> bit) & 1u) {
      uint32_t r = wpre[w] + __popc(keepw[w] & ((1u << bit) - 1u));
      if (r < MAXDET) {
        float4 bx = topBox[(size_t)b * KSEL + i];
        float sc = topScore[(size_t)b * KSEL + i];
        float* o = out + ((size_t)b * MAXDET + r) * 6;
        o[0] = bx.x; o[1] = bx.y; o[2] = bx.z; o[3] = bx.w;
        o[4] = sc;   o[5] = 0.0f;
      }
    }
  }
}

// ---------------------------------------------------------------------------
extern "C" void kernel_launch(void* const* d_in, const int* in_sizes, int n_in,
                              void* d_out, int out_size, void* d_ws, size_t ws_size,
                              hipStream_t stream) {
  (void)in_sizes; (void)n_in; (void)out_size; (void)ws_size;
  const float* x = (const float*)d_in[0];
  float* out = (float*)d_out;
  char* ws = (char*)d_ws;

  // workspace layout (all 16B-aligned offsets), ~11.3 MB total
  unsigned long long* keys = (unsigned long long*)ws;                     // 32*25200*8  = 6,451,200
  float*    topScore = (float*)(ws + 6451200);                            // 32*1024*4   =   131,072
  float4*   topBox   = (float4*)(ws + 6451200 + 131072);                  // 32*1024*16  =   524,288
  uint32_t* mask     = (uint32_t*)(ws + 6451200 + 131072 + 524288);       // 32*1024*32*4= 4,194,304

  k_init<<<4096, 256, 0, stream>>>(x, out, keys, mask);
  k_select<<<BATCH, 1024, 0, stream>>>(x, keys, topScore, topBox);
  k_iou<<<(BATCH * 4096) / 8, 256, 0, stream>>>(topBox, mask);  // 1 wave32 / 16x16 tile
  k_nms<<<BATCH, 1024, 0, stream>>>(topScore, topBox, mask, out);
}